// SequenceEmbedding_16647293239442
// MI455X (gfx1250) — compile-verified
//
#include <hip/hip_runtime.h>

#define L 4096
#define ROWS_PER_BLOCK 16
#define THREADS 256

typedef float float4_t __attribute__((ext_vector_type(4)));

__global__ __launch_bounds__(THREADS)
void seq_embed_kernel(const float* __restrict__ one_hot, float* __restrict__ out) {
    // 64 KB staged copy of the full one_hot table [L][4]
    __shared__ float s_oh[L * 4];
    // 16 KB transposed column vector for the "col" channels
    __shared__ float s_col[L];

    const int tid = threadIdx.x;
    const int bid = blockIdx.x;
    const int c   = bid >> 8;                      // channel 0..7
    const int i0  = (bid & 255) * ROWS_PER_BLOCK;  // row-strip start

    // ---- Stage one_hot (65536 B) into LDS via CDNA5 async global->LDS DMA ----
    // 256 threads x 16 iterations x 16 B = 65536 B. Tracked on ASYNCcnt.
    {
        uint32_t lds_base = (uint32_t)(uintptr_t)(&s_oh[0]); // low 32 bits of generic LDS ptr = LDS offset
        const char* gbase = (const char*)one_hot;
        #pragma unroll
        for (int k = 0; k < (L * 4 * 4) / (THREADS * 16); ++k) {
            uint32_t byteoff = (uint32_t)(k * THREADS + tid) * 16u;
            uint32_t ldsa    = lds_base + byteoff;
            const char* ga   = gbase + byteoff;
            asm volatile("global_load_async_to_lds_b128 %0, %1, off"
                         :: "v"(ldsa), "v"(ga)
                         : "memory");
        }
        asm volatile("s_wait_asynccnt 0" ::: "memory");
    }
    __syncthreads();

    const bool isRow = (c < 4);
    const int  cc    = isRow ? c : (c - 4);

    // ---- For col channels, transpose the needed column into contiguous LDS ----
    if (!isRow) {
        #pragma unroll
        for (int j = tid; j < L; j += THREADS)
            s_col[j] = s_oh[j * 4 + cc];
    }
    __syncthreads();

    // ---- Streaming store phase: 16 rows x 16 KB, non-temporal b128 stores ----
    // Per row: 1024 float4 / 256 threads = 4 float4 per thread.
    for (int r = 0; r < ROWS_PER_BLOCK; ++r) {
        const int i = i0 + r;
        float4_t* rowp = (float4_t*)(out + ((size_t)c * L + (size_t)i) * L);
        if (isRow) {
            const float v = s_oh[i * 4 + cc];
            const float4_t vv = {v, v, v, v};
            #pragma unroll
            for (int it = 0; it < 4; ++it)
                __builtin_nontemporal_store(vv, rowp + it * THREADS + tid);
        } else {
            #pragma unroll
            for (int it = 0; it < 4; ++it) {
                const int j4 = it * THREADS + tid;                 // float4 index within row
                const float4_t vv = *(const float4_t*)&s_col[j4 * 4]; // ds_load_b128, conflict-free
                __builtin_nontemporal_store(vv, rowp + j4);
            }
        }
    }
}

extern "C" void kernel_launch(void* const* d_in, const int* in_sizes, int n_in,
                              void* d_out, int out_size, void* d_ws, size_t ws_size,
                              hipStream_t stream) {
    (void)in_sizes; (void)n_in; (void)d_ws; (void)ws_size; (void)out_size;
    const float* one_hot = (const float*)d_in[0];
    float* out = (float*)d_out;
    // 8 channels x (4096 / 16) row strips = 2048 blocks of 256 threads (8 wave32s)
    dim3 grid(8 * (L / ROWS_PER_BLOCK));
    seq_embed_kernel<<<grid, THREADS, 0, stream>>>(one_hot, out);
}